// GATLayer_74148315398470
// MI455X (gfx1250) — compile-verified
//
#include <hip/hip_runtime.h>

typedef __attribute__((ext_vector_type(2))) float v2f;
typedef __attribute__((ext_vector_type(8))) float v8f;

#define GAT_HEADS 4
#define GAT_INF   128
#define GAT_OUTC  128    // HEADS * OUT_F
#define XS_STRIDE 132    // x tile LDS row stride (dwords): row step = 4 banks
#define WT2_STRIDE 288   // K-pair interleaved W row stride (dwords): 288%64==32
                         // -> lanes 16-31 (next pair-row) hit banks +32..+63

__device__ __forceinline__ float gat_lrelu(float v) { return v > 0.f ? v : 0.2f * v; }

// Monotone float<->uint mapping so float max == unsigned max (handles negatives).
__device__ __forceinline__ unsigned gat_fenc(float f) {
  unsigned u = __float_as_uint(f);
  return (u & 0x80000000u) ? ~u : (u | 0x80000000u);
}
__device__ __forceinline__ float gat_fdec(unsigned u) {
  return __uint_as_float((u & 0x80000000u) ? (u & 0x7FFFFFFFu) : ~u);
}

// ---------------------------------------------------------------------------
// Kernel 0: zero d_out + denom, seed global max with encode(-inf) = 0x007FFFFF
// ---------------------------------------------------------------------------
__global__ void gat_init(float* __restrict__ out, float* __restrict__ denom,
                         unsigned* __restrict__ gmax, int nOut, int nDen) {
  int i = blockIdx.x * blockDim.x + threadIdx.x;
  if (i == 0) *gmax = 0x007FFFFFu;  // gat_fenc(-inf)
  if (i < nOut) out[i] = 0.f;
  if (i < nDen) denom[i] = 0.f;
}

// ---------------------------------------------------------------------------
// Kernel 1: proj = x @ W  via V_WMMA_F32_16X16X4_F32 (exact f32 path)
// Block = 256 threads = 8 waves; block computes a 128x128 tile of proj.
// A 16x4 f32 fragment: lanes 0-15 -> (M=lane, K=k..k+1), lanes 16-31 -> K=k+2..k+3
// B 4x16 f32 fragment: lane&15 = N column, same half-wave K split.
// C/D 16x16 f32: VGPR r -> M = r + (lane<16 ? 0 : 8), N = lane&15.
// W staged K-pair interleaved: wt2[k>>1][c] = {W[k][c], W[k+1][c]} -> every
// fragment operand is ONE aligned ds_load_b64, conflict-free across 64 banks.
// All 16 B fragments of a step preloaded into independent regs so the
// scheduler can batch the LDS loads behind a single dscnt wait.
// ---------------------------------------------------------------------------
__global__ __launch_bounds__(256) void gat_gemm_wmma(
    const float* __restrict__ x, const float* __restrict__ W,
    float* __restrict__ proj, int nNodes) {
  extern __shared__ float smem[];
  float* xs  = smem;                    // [128][XS_STRIDE]
  float* wt2 = smem + 128 * XS_STRIDE;  // [64][WT2_STRIDE] (pairs, 2 dw per col)
  const int tid  = threadIdx.x;
  const int row0 = blockIdx.x * 128;

  // Stage W (128x128 f32) into K-pair interleaved LDS layout.
  for (int i = tid; i < 64 * 32; i += 256) {
    int r2 = i >> 5;            // k pair index
    int c  = (i & 31) << 2;     // column chunk (4 cols)
    float4 e = *(const float4*)(W + (2 * r2 + 0) * GAT_OUTC + c);
    float4 o = *(const float4*)(W + (2 * r2 + 1) * GAT_OUTC + c);
    float* dstp = wt2 + r2 * WT2_STRIDE + 2 * c;
    float4 p0 = {e.x, o.x, e.y, o.y};
    float4 p1 = {e.z, o.z, e.w, o.w};
    *(float4*)(dstp + 0) = p0;
    *(float4*)(dstp + 4) = p1;
  }
  // Stage this block's 128-row x tile (zero-padded past nNodes).
  for (int i = tid; i < 128 * 32; i += 256) {
    int r = i >> 5;
    int c = (i & 31) << 2;
    int gr = row0 + r;
    float4 v = {0.f, 0.f, 0.f, 0.f};
    if (gr < nNodes) v = *(const float4*)(x + (size_t)gr * GAT_INF + c);
    *(float4*)(xs + r * XS_STRIDE + c) = v;
  }
  __syncthreads();

  const int wave  = tid >> 5;   // wave32
  const int lane  = tid & 31;
  const int l16   = lane & 15;
  const int lhalf = lane >> 4;  // 0: K+0..1, 1: K+2..3
  const int mrow  = wave * 16;  // wave's 16-row tile inside the block

  const v8f vzero = {0.f, 0.f, 0.f, 0.f, 0.f, 0.f, 0.f, 0.f};
  v8f acc[8];
#pragma unroll
  for (int n = 0; n < 8; ++n) acc[n] = vzero;

  const float* arow = xs + (mrow + l16) * XS_STRIDE + lhalf * 2;  // + 2*k2
  const float* brow = wt2 + (size_t)lhalf * WT2_STRIDE + 2 * l16; // + k2*WT2_STRIDE + n*32

  for (int k2 = 0; k2 < 64; k2 += 2) {  // two K-pair rows (K step 4) per iter
    v2f a0 = *(const v2f*)(arow + 2 * k2);
    v2f a1 = *(const v2f*)(arow + 2 * k2 + 4);
    const float* bk0 = brow + (size_t)k2 * WT2_STRIDE;
    const float* bk1 = bk0 + WT2_STRIDE;
    v2f b0[8], b1[8];
#pragma unroll
    for (int n = 0; n < 8; ++n) b0[n] = *(const v2f*)(bk0 + n * 32);
#pragma unroll
    for (int n = 0; n < 8; ++n) b1[n] = *(const v2f*)(bk1 + n * 32);
#pragma unroll
    for (int n = 0; n < 8; ++n)
      acc[n] = __builtin_amdgcn_wmma_f32_16x16x4_f32(
          false, a0, false, b0[n], (short)0, acc[n], false, false);
#pragma unroll
    for (int n = 0; n < 8; ++n)
      acc[n] = __builtin_amdgcn_wmma_f32_16x16x4_f32(
          false, a1, false, b1[n], (short)0, acc[n], false, false);
  }

  const int rbase = row0 + mrow + lhalf * 8;
  if (row0 + 128 <= nNodes) {
    // Block-uniform fast path: no per-row guards.
#pragma unroll
    for (int n = 0; n < 8; ++n) {
      float* op = proj + (size_t)rbase * GAT_OUTC + n * 16 + l16;
#pragma unroll
      for (int r = 0; r < 8; ++r) op[(size_t)r * GAT_OUTC] = acc[n][r];
    }
  } else {
#pragma unroll
    for (int n = 0; n < 8; ++n) {
#pragma unroll
      for (int r = 0; r < 8; ++r) {
        int gr = rbase + r;
        if (gr < nNodes) proj[(size_t)gr * GAT_OUTC + n * 16 + l16] = acc[n][r];
      }
    }
  }
}

// ---------------------------------------------------------------------------
// Kernel 2: per-(node,head) attention score dots
// ---------------------------------------------------------------------------
__global__ void gat_scores(const float* __restrict__ proj,
                           const float* __restrict__ att_src,
                           const float* __restrict__ att_dst,
                           float* __restrict__ s_src, float* __restrict__ s_dst,
                           int nNodes) {
  int idx = blockIdx.x * blockDim.x + threadIdx.x;
  if (idx >= nNodes * GAT_HEADS) return;
  int n = idx >> 2, h = idx & 3;
  const float* p  = proj + (size_t)n * GAT_OUTC + h * 32;
  const float* as = att_src + h * 32;
  const float* ad = att_dst + h * 32;
  float ss = 0.f, sd = 0.f;
#pragma unroll
  for (int f = 0; f < 32; f += 4) {
    float4 pv = *(const float4*)(p + f);
    float4 av = *(const float4*)(as + f);
    float4 dv = *(const float4*)(ad + f);
    ss += pv.x * av.x + pv.y * av.y + pv.z * av.z + pv.w * av.w;
    sd += pv.x * dv.x + pv.y * dv.y + pv.z * dv.z + pv.w * dv.w;
  }
  s_src[idx] = ss;
  s_dst[idx] = sd;
}

// ---------------------------------------------------------------------------
// Kernel 3: global max of leaky_relu(e) over all edges/heads
// ---------------------------------------------------------------------------
__global__ __launch_bounds__(256) void gat_edge_max(
    const int* __restrict__ src, const int* __restrict__ dst,
    const float* __restrict__ s_src, const float* __restrict__ s_dst,
    unsigned* __restrict__ gmax, int nEdges) {
  int e = blockIdx.x * blockDim.x + threadIdx.x;
  float m = -__builtin_inff();
  if (e < nEdges) {
    int s = src[e], d = dst[e];
    float4 a = *(const float4*)(s_src + (size_t)s * 4);
    float4 b = *(const float4*)(s_dst + (size_t)d * 4);
    m = fmaxf(fmaxf(gat_lrelu(a.x + b.x), gat_lrelu(a.y + b.y)),
              fmaxf(gat_lrelu(a.z + b.z), gat_lrelu(a.w + b.w)));
  }
#pragma unroll
  for (int o = 16; o > 0; o >>= 1) m = fmaxf(m, __shfl_xor(m, o, 32));
  __shared__ float sm[8];
  if ((threadIdx.x & 31) == 0) sm[threadIdx.x >> 5] = m;
  __syncthreads();
  if (threadIdx.x == 0) {
    float bm = sm[0];
#pragma unroll
    for (int w = 1; w < 8; ++w) bm = fmaxf(bm, sm[w]);
    atomicMax(gmax, gat_fenc(bm));
  }
}

// ---------------------------------------------------------------------------
// Kernel 4: softmax denominator scatter (recomputes e; no exp_e buffer needed)
// ---------------------------------------------------------------------------
__global__ void gat_edge_denom(const int* __restrict__ src,
                               const int* __restrict__ dst,
                               const float* __restrict__ s_src,
                               const float* __restrict__ s_dst,
                               const unsigned* __restrict__ gmax,
                               float* __restrict__ denom, int nEdges) {
  int e = blockIdx.x * blockDim.x + threadIdx.x;
  if (e >= nEdges) return;
  float gm = gat_fdec(*gmax);
  int s = src[e], d = dst[e];
  float4 a = *(const float4*)(s_src + (size_t)s * 4);
  float4 b = *(const float4*)(s_dst + (size_t)d * 4);
  atomicAdd(&denom[(size_t)d * 4 + 0], __expf(gat_lrelu(a.x + b.x) - gm));
  atomicAdd(&denom[(size_t)d * 4 + 1], __expf(gat_lrelu(a.y + b.y) - gm));
  atomicAdd(&denom[(size_t)d * 4 + 2], __expf(gat_lrelu(a.z + b.z) - gm));
  atomicAdd(&denom[(size_t)d * 4 + 3], __expf(gat_lrelu(a.w + b.w) - gm));
}

// ---------------------------------------------------------------------------
// Kernel 5: wave-per-edge weighted scatter-add. Lane owns 4 contiguous floats
// (head = lane/8); one float4 gather of proj[src], 4 f32 atomic adds to out[dst].
// Working set is L2-resident (proj+out ~102 MB << 192 MB L2).
// ---------------------------------------------------------------------------
__global__ __launch_bounds__(256) void gat_edge_aggr(
    const int* __restrict__ src, const int* __restrict__ dst,
    const float* __restrict__ s_src, const float* __restrict__ s_dst,
    const float* __restrict__ denom, const unsigned* __restrict__ gmax,
    const float* __restrict__ proj, float* __restrict__ out, int nEdges) {
  long long gtid = (long long)blockIdx.x * blockDim.x + threadIdx.x;
  int e = (int)(gtid >> 5);
  int lane = threadIdx.x & 31;
  if (e >= nEdges) return;
  int s = src[e], d = dst[e];
  int h = lane >> 3;
  float gm = gat_fdec(*gmax);
  float ev = gat_lrelu(s_src[(size_t)s * 4 + h] + s_dst[(size_t)d * 4 + h]);
  float alpha = __expf(ev - gm) / (denom[(size_t)d * 4 + h] + 1e-16f);
  float4 pv = *(const float4*)(proj + (size_t)s * GAT_OUTC + lane * 4);
  float* o = out + (size_t)d * GAT_OUTC + lane * 4;
  atomicAdd(o + 0, pv.x * alpha);
  atomicAdd(o + 1, pv.y * alpha);
  atomicAdd(o + 2, pv.z * alpha);
  atomicAdd(o + 3, pv.w * alpha);
}

// ---------------------------------------------------------------------------
extern "C" void kernel_launch(void* const* d_in, const int* in_sizes, int n_in,
                              void* d_out, int out_size, void* d_ws, size_t ws_size,
                              hipStream_t stream) {
  (void)n_in; (void)out_size; (void)ws_size;
  const float* x       = (const float*)d_in[0];
  const int*   adj     = (const int*)d_in[1];
  const float* W       = (const float*)d_in[2];
  const float* att_src = (const float*)d_in[3];
  const float* att_dst = (const float*)d_in[4];

  const int nNodes = in_sizes[0] / GAT_INF;
  const int nEdges = in_sizes[1] / 2;
  const int* src = adj;
  const int* dst = adj + nEdges;
  float* out = (float*)d_out;

  // Scratch layout (256B aligned): gmax | denom[N*H] | s_src[N*H] | s_dst[N*H] | proj[N*128]
  char* wsb = (char*)d_ws;
  unsigned* gmax = (unsigned*)wsb;
  size_t off = 256;
  size_t nhBytes = (((size_t)nNodes * GAT_HEADS * sizeof(float)) + 255) & ~(size_t)255;
  float* denom = (float*)(wsb + off); off += nhBytes;
  float* s_src = (float*)(wsb + off); off += nhBytes;
  float* s_dst = (float*)(wsb + off); off += nhBytes;
  float* proj  = (float*)(wsb + off);  // nNodes * 128 floats (51.2 MB)

  const int nOut = nNodes * GAT_OUTC;
  const int nDen = nNodes * GAT_HEADS;

  gat_init<<<(nOut + 255) / 256, 256, 0, stream>>>(out, denom, gmax, nOut, nDen);

  const size_t smemBytes =
      (size_t)(128 * XS_STRIDE + 64 * WT2_STRIDE) * sizeof(float);
  gat_gemm_wmma<<<(nNodes + 127) / 128, 256, smemBytes, stream>>>(x, W, proj, nNodes);

  gat_scores<<<(nNodes * GAT_HEADS + 255) / 256, 256, 0, stream>>>(
      proj, att_src, att_dst, s_src, s_dst, nNodes);

  gat_edge_max<<<(nEdges + 255) / 256, 256, 0, stream>>>(src, dst, s_src, s_dst, gmax, nEdges);

  gat_edge_denom<<<(nEdges + 255) / 256, 256, 0, stream>>>(
      src, dst, s_src, s_dst, gmax, denom, nEdges);

  long long aggrThreads = (long long)nEdges * 32;
  gat_edge_aggr<<<(unsigned)((aggrThreads + 255) / 256), 256, 0, stream>>>(
      src, dst, s_src, s_dst, denom, gmax, proj, out, nEdges);
}